// TpsGridGen_41068477284473
// MI455X (gfx1250) — compile-verified
//
#include <hip/hip_runtime.h>

#define GRID_N 25
#define KDIM   28          // 25 RBF + [1, gx, gy]
#define KPAD   29          // LDS row stride (odd -> conflict-free)
#define OUT_H  256
#define OUT_W  192
#define NPIX   (OUT_H * OUT_W)
#define BATCH  64

typedef __attribute__((ext_vector_type(2))) float v2f;
typedef __attribute__((ext_vector_type(8))) float v8f;

// ---------------------------------------------------------------------------
// Prep: wgt[coord][b][j] = sum_m Li[j,m] * (theta[b, coord*25 + m] + P[m])
// j = 0..27 covers both nonlinear weights W (j<25) and affine A (j=25..27),
// ordered to match the feature vector [U_0..U_24, 1, gx, gy].
// 3584 outputs, 25-MAC dot each.
// ---------------------------------------------------------------------------
__global__ __launch_bounds__(256) void tps_prep(
    const float* __restrict__ theta, const float* __restrict__ Px,
    const float* __restrict__ Py, const float* __restrict__ Li,
    float* __restrict__ wgt)
{
    int idx = threadIdx.x + blockIdx.x * 256;
    if (idx >= 2 * BATCH * KDIM) return;
    int coord = idx / (BATCH * KDIM);
    int r     = idx - coord * (BATCH * KDIM);
    int b     = r / KDIM;
    int j     = r - b * KDIM;
    const float* P    = coord ? Py : Px;
    const float* th   = theta + b * (2 * GRID_N) + coord * GRID_N;
    const float* Lrow = Li + j * KDIM;
    float s = 0.f;
#pragma unroll
    for (int m = 0; m < GRID_N; ++m)
        s = fmaf(Lrow[m], th[m] + P[m], s);
    wgt[idx] = s;
}

// ---------------------------------------------------------------------------
// Main: per block, 32 pixels x 64 batches.
//   D[16 batch x 16 pixel] = A(weights 16x28) * B(features 28x16), K in 7x4.
//   accX/accY accumulated with V_WMMA_F32_16X16X4_F32.
// Stores are float2 {Xp,Yp}: lanes 0-15 write 16 consecutive pixels of one
// batch row -> contiguous 128B global_store_b64 runs (store-BW bound kernel).
// ---------------------------------------------------------------------------
__global__ __launch_bounds__(256) void tps_main(
    const float* __restrict__ gx, const float* __restrict__ gy,
    const float* __restrict__ Px, const float* __restrict__ Py,
    const float* __restrict__ wgt,   // [2][64][28]
    float* __restrict__ out)         // [64][H][W][2]
{
    __shared__ float lds_w[2 * BATCH * KPAD];  // weights, x then y
    __shared__ float lds_f[32 * KPAD];         // feature tile

    const int tid = threadIdx.x;
    const int pixBase = blockIdx.x * 32;

    // Stage weights (restride 28 -> 29)
    for (int i = tid; i < 2 * BATCH * KDIM; i += 256) {
        int cb = i / KDIM;          // coord*64 + batch
        int k  = i - cb * KDIM;
        lds_w[cb * KPAD + k] = wgt[i];
    }
    // Compute features once per (pixel, k); log evaluated 25x per pixel total.
    for (int i = tid; i < 32 * KDIM; i += 256) {
        int p = i / KDIM;
        int k = i - p * KDIM;
        int gp = pixBase + p;
        float xv = gx[gp], yv = gy[gp];
        float f;
        if (k < GRID_N) {
            float dx = xv - Px[k];
            float dy = yv - Py[k];
            float d2 = dx * dx + dy * dy;
            d2 = (d2 == 0.f) ? 1.f : d2;   // U(0) = 0 convention
            f = d2 * __logf(d2);
        } else {
            f = (k == GRID_N) ? 1.f : ((k == GRID_N + 1) ? xv : yv);
        }
        lds_f[p * KPAD + k] = f;
    }
    __syncthreads();

    const int lane = tid & 31;
    const int wid  = tid >> 5;
    const int batchBase = (wid & 3) * 16;   // 4 batch tiles cover 64 batches
    const int pixLocal  = (wid >> 2) * 16;  // 2 pixel tiles cover 32 pixels
    const int hl   = lane & 15;
    const int kOff = (lane >> 4) * 2;       // lanes 16-31 hold K+2, K+3

    const float* wx = &lds_w[(batchBase + hl) * KPAD];
    const float* wy = &lds_w[(BATCH + batchBase + hl) * KPAD];
    const float* ft = &lds_f[(pixLocal + hl) * KPAD];

    v8f accX = {};
    v8f accY = {};
#pragma unroll
    for (int k = 0; k < 7; ++k) {           // K = 28 = 7 x 4
        int kb = 4 * k + kOff;
        v2f a_x = { wx[kb], wx[kb + 1] };   // A: 16x4 f32 (weights, M=batch)
        v2f a_y = { wy[kb], wy[kb + 1] };
        v2f b_f = { ft[kb], ft[kb + 1] };   // B: 4x16 f32 (features, N=pixel)
        accX = __builtin_amdgcn_wmma_f32_16x16x4_f32(
            false, a_x, false, b_f, (short)0, accX, false, false);
        accY = __builtin_amdgcn_wmma_f32_16x16x4_f32(
            false, a_y, false, b_f, (short)0, accY, false, false);
    }

    // D layout: VGPR j -> batch row = batchBase + j + (lane>=16 ? 8 : 0),
    //           pixel col = pixBase + pixLocal + (lane & 15)
    const int gpix = pixBase + pixLocal + hl;
    const int bAdd = batchBase + ((lane >> 4) << 3);
    float2* o2 = (float2*)out;              // [b][pix] of (Xp, Yp)
#pragma unroll
    for (int j = 0; j < 8; ++j) {
        int b = bAdd + j;
        o2[(size_t)b * NPIX + gpix] = make_float2(accX[j], accY[j]);
    }
}

// ---------------------------------------------------------------------------
extern "C" void kernel_launch(void* const* d_in, const int* in_sizes, int n_in,
                              void* d_out, int out_size, void* d_ws, size_t ws_size,
                              hipStream_t stream) {
    const float* theta = (const float*)d_in[0];
    const float* gx    = (const float*)d_in[1];
    const float* gy    = (const float*)d_in[2];
    const float* Px    = (const float*)d_in[3];
    const float* Py    = (const float*)d_in[4];
    const float* Li    = (const float*)d_in[5];
    float* wgt = (float*)d_ws;     // 2*64*28 floats = 14336 B
    float* out = (float*)d_out;

    hipLaunchKernelGGL(tps_prep, dim3(14), dim3(256), 0, stream,
                       theta, Px, Py, Li, wgt);
    hipLaunchKernelGGL(tps_main, dim3(NPIX / 32), dim3(256), 0, stream,
                       gx, gy, Px, Py, wgt, out);
}